// LengthRegulator_90280212562587
// MI455X (gfx1250) — compile-verified
//
#include <hip/hip_runtime.h>
#include <hip/hip_bf16.h>
#include <stdint.h>

// Problem constants (match reference harness)
#define B_ 32
#define S_ 512
#define D_ 512
#define T_ 2048
#define SLOWDOWN_ 0.5f

// ---------------------------------------------------------------------------
// Phase 1: per-batch duration quantization + inclusive scan.
// One 512-thread block per batch row. Writes:
//   - d (int32)  -> tail of d_out (concatenated tuple output, bit-exact int32)
//   - cumsum     -> workspace (consumed by phase 2)
// ---------------------------------------------------------------------------
__global__ __launch_bounds__(S_) void lr_scan_kernel(const float* __restrict__ dur,
                                                     int* __restrict__ d_out,
                                                     int* __restrict__ csum) {
    __shared__ int sm[S_];
    const int b = blockIdx.x;
    const int s = threadIdx.x;

    // d = int(max(dur + 0.5, 1.0)) ; values are >= 1 so truncation == .int()
    const float f = dur[b * S_ + s] + SLOWDOWN_;
    const int d = (int)fmaxf(f, 1.0f);

    sm[s] = d;
    __syncthreads();

    // Hillis-Steele inclusive scan over 512 elements (9 steps)
    #pragma unroll
    for (int off = 1; off < S_; off <<= 1) {
        const int v = (s >= off) ? sm[s - off] : 0;
        __syncthreads();
        sm[s] += v;
        __syncthreads();
    }

    csum[b * S_ + s]  = sm[s];
    d_out[b * S_ + s] = d;
}

// ---------------------------------------------------------------------------
// Phase 2: frame gather via CDNA5 async global<->LDS DMA.
// One 128-thread block per (b, t) frame. Each thread moves one float4 (16 B)
// of the 2 KB row: global_load_async_to_lds_b128 -> wait asynccnt ->
// global_store_async_from_lds_b128. No VGPR staging of the payload.
// ---------------------------------------------------------------------------
__global__ __launch_bounds__(128) void lr_gather_kernel(const float* __restrict__ seq,
                                                        const int* __restrict__ csum,
                                                        float* __restrict__ out) {
    __shared__ float4 buf[128];

    const int bt = blockIdx.x;          // 0 .. B*T-1
    const int b  = bt >> 11;            // T_ == 2048
    const int t  = bt & (T_ - 1);

    // idx = #(csum[b,:] <= t)  == searchsorted(csum, t, side='right')
    const int* __restrict__ c = csum + b * S_;
    int lo = 0, hi = S_;
    while (lo < hi) {
        const int mid = (lo + hi) >> 1;
        if (c[mid] <= t) lo = mid + 1; else hi = mid;
    }

    const int tid = threadIdx.x;
    float4* dst = reinterpret_cast<float4*>(out) + (size_t)bt * (D_ / 4) + tid;

    if (lo < S_) {
        const float4* src =
            reinterpret_cast<const float4*>(seq) + ((size_t)b * S_ + lo) * (D_ / 4) + tid;

        // Raw 32-bit LDS offset for the async engine (VDST/VSRC hold LDS addr).
        typedef __attribute__((address_space(3))) float4 lds_float4;
        const unsigned lds = (unsigned)(uintptr_t)(lds_float4*)&buf[tid];

        // Memory -> LDS (ASYNCcnt++)
        asm volatile("global_load_async_to_lds_b128 %0, %1, off"
                     :: "v"(lds), "v"(src) : "memory");
        // Load must land in LDS before the store engine reads it back.
        asm volatile("s_wait_asynccnt 0" ::: "memory");
        // LDS -> Memory (ASYNCcnt++)
        asm volatile("global_store_async_from_lds_b128 %0, %1, off"
                     :: "v"(dst), "v"(lds) : "memory");
        asm volatile("s_wait_asynccnt 0" ::: "memory");
    } else {
        // Padding frame: zero-fill directly (16 B store per lane).
        *dst = make_float4(0.f, 0.f, 0.f, 0.f);
    }
}

// ---------------------------------------------------------------------------
// Entry point
// ---------------------------------------------------------------------------
extern "C" void kernel_launch(void* const* d_in, const int* in_sizes, int n_in,
                              void* d_out, int out_size, void* d_ws, size_t ws_size,
                              hipStream_t stream) {
    (void)in_sizes; (void)n_in; (void)out_size; (void)ws_size;

    const float* seq = (const float*)d_in[0];   // [B,S,D] float32
    const float* dur = (const float*)d_in[1];   // [B,S]   float32
    // d_in[2] = max_mel_length scalar (== T_, baked in)

    float* out  = (float*)d_out;                               // [B,T,D] float32
    int*   dint = (int*)(out + (size_t)B_ * T_ * D_);          // [B,S] int32 (tuple tail)
    int*   csum = (int*)d_ws;                                  // [B,S] int32 scratch

    lr_scan_kernel<<<dim3(B_), dim3(S_), 0, stream>>>(dur, dint, csum);
    lr_gather_kernel<<<dim3(B_ * T_), dim3(128), 0, stream>>>(seq, csum, out);
}